// ProjectToPlane_31971736551440
// MI455X (gfx1250) — compile-verified
//
#include <hip/hip_runtime.h>
#include <stdint.h>

#define H 512
#define W 512
#define NPIX (H * W)
#define BLOCK 256
#define PTS_PER_TILE 2048
#define FLOATS_PER_TILE (PTS_PER_TILE * 3)                 // 6144 floats = 24KB per buffer
#define B128_PER_THREAD (FLOATS_PER_TILE / (BLOCK * 4))    // 6 async b128 instrs per wave/tile

// ---- monotone float<->uint key mapping (integer order == float order) ----
__device__ __forceinline__ uint32_t f2key(float f) {
    uint32_t b = __float_as_uint(f);
    return (b & 0x80000000u) ? ~b : (b | 0x80000000u);
}
__device__ __forceinline__ float key2f(uint32_t k) {
    uint32_t b = (k & 0x80000000u) ? (k ^ 0x80000000u) : ~k;
    return __uint_as_float(b);
}

// mimic jnp.searchsorted(arange(-256,256), v, 'right') - 1 followed by JAX
// negative-index wrap in .at[] scatter: floor(v)+256, clamp high, -1 wraps to hi.
__device__ __forceinline__ int bin_of(float v, int hi) {
    int b = (int)floorf(v) + 256;
    b = (b > hi) ? hi : b;
    b = (b < 0) ? hi : b;
    return b;
}

// ---------------------------------------------------------------- init ----
__global__ void init_ws(uint32_t* __restrict__ minmax,
                        float* __restrict__ sums,
                        float* __restrict__ counts) {
    int i = blockIdx.x * blockDim.x + threadIdx.x;
    if (i == 0) { minmax[0] = 0xFFFFFFFFu; minmax[1] = 0u; }
    int stride = gridDim.x * blockDim.x;
    for (int p = i; p < NPIX; p += stride) {
        sums[p]   = 0.0f;
        counts[p] = 0.0f;
    }
}

// issue one tile's worth of async Global->LDS b128 copies (6 instrs per wave),
// non-temporal so the 100MB point stream doesn't evict the L2-resident histogram.
__device__ __forceinline__ void issue_tile_async(const float* __restrict__ pc,
                                                 long long baseFloat,
                                                 uint32_t ldsBufBase, int tid) {
    #pragma unroll
    for (int j = 0; j < B128_PER_THREAD; ++j) {
        int fofs = (j * BLOCK + tid) * 4;  // float offset in tile (16B aligned)
        uint32_t lds_addr = ldsBufBase + (uint32_t)(fofs * 4);
        uint64_t gaddr = (uint64_t)(uintptr_t)(pc + baseFloat + fofs);
        asm volatile("global_load_async_to_lds_b128 %0, %1, off th:TH_LOAD_NT"
                     :: "v"(lds_addr), "v"(gaddr)
                     : "memory");
    }
}

// --------------------- single fused pass: bin + raw-z scatter + z min/max ---
// z-normalization is affine and the scatter-add is linear, so it commutes with
// accumulation; we accumulate RAW z and apply 255*(mean - zmin)/range at the end.
// This reads the 100MB point stream exactly ONCE (the HBM roofline floor).
__global__ void __launch_bounds__(BLOCK) bin_points(
    const float* __restrict__ pc, long long n,
    const float* __restrict__ C7, const float* __restrict__ DR,
    const float* __restrict__ DL, uint32_t* __restrict__ minmax,
    float* __restrict__ sums, float* __restrict__ counts)
{
    __shared__ __align__(16) float tile[2][FLOATS_PER_TILE];  // 48KB, double buffer

    // factor = 0.75 * H / ||C7 - (DR + DL/2)||
    float dx = C7[0] - (DR[0] + 0.5f * DL[0]);
    float dy = C7[1] - (DR[1] + 0.5f * DL[1]);
    float dz = C7[2] - (DR[2] + 0.5f * DL[2]);
    float factor = (0.75f * (float)H) / sqrtf(dx * dx + dy * dy + dz * dz);

    const int tid = threadIdx.x;
    const uint32_t ldsBase0 = (uint32_t)(uintptr_t)(&tile[0][0]);
    const uint32_t ldsBase1 = (uint32_t)(uintptr_t)(&tile[1][0]);

    uint32_t kmin = 0xFFFFFFFFu, kmax = 0u;   // running z min/max keys

    long long nFull  = n / PTS_PER_TILE;
    long long stride = gridDim.x;
    long long t0     = blockIdx.x;

    // ---- depth-2 async pipeline prologue ----
    if (t0 < nFull)
        issue_tile_async(pc, t0 * (long long)FLOATS_PER_TILE, ldsBase0, tid);
    if (t0 + stride < nFull)
        issue_tile_async(pc, (t0 + stride) * (long long)FLOATS_PER_TILE, ldsBase1, tid);

    int buf = 0;
    for (long long t = t0; t < nFull; t += stride, buf ^= 1) {
        // batches retire in order: allow the newer batch (6 instrs) to stay in flight
        if (t + stride < nFull)
            asm volatile("s_wait_asynccnt 6" ::: "memory");
        else
            asm volatile("s_wait_asynccnt 0" ::: "memory");
        __syncthreads();  // (a) every wave's data for `buf` has landed

        const float* __restrict__ tp = &tile[buf][0];
        #pragma unroll 4
        for (int p = tid; p < PTS_PER_TILE; p += BLOCK) {
            float x = factor * tp[3 * p + 0];
            float y = factor * tp[3 * p + 1];
            float z = tp[3 * p + 2];
            uint32_t k = f2key(z);
            kmin = kmin < k ? kmin : k;
            kmax = kmax > k ? kmax : k;
            int pix = bin_of(y, H - 1) * W + bin_of(x, W - 1);
            atomicAdd(&sums[pix], z);      // global_atomic_add_f32, L2-resident table
            atomicAdd(&counts[pix], 1.0f);
        }
        __syncthreads();  // (b) all waves done reading `buf` before refilling it

        long long t2 = t + 2 * stride;
        if (t2 < nFull)
            issue_tile_async(pc, t2 * (long long)FLOATS_PER_TILE,
                             buf ? ldsBase1 : ldsBase0, tid);
    }

    // --- tail points straight from global (dead for N=8388608, kept for safety) ---
    if (blockIdx.x == 0) {
        for (long long p = nFull * PTS_PER_TILE + tid; p < n; p += BLOCK) {
            float x = factor * __builtin_nontemporal_load(&pc[3 * p + 0]);
            float y = factor * __builtin_nontemporal_load(&pc[3 * p + 1]);
            float z = __builtin_nontemporal_load(&pc[3 * p + 2]);
            uint32_t k = f2key(z);
            kmin = kmin < k ? kmin : k;
            kmax = kmax > k ? kmax : k;
            int pix = bin_of(y, H - 1) * W + bin_of(x, W - 1);
            atomicAdd(&sums[pix], z);
            atomicAdd(&counts[pix], 1.0f);
        }
    }

    // ---- fold per-lane z min/max into global: wave32 butterfly + 2 atomics/wave ----
    #pragma unroll
    for (int off = 16; off > 0; off >>= 1) {
        uint32_t a = (uint32_t)__shfl_xor((int)kmin, off, 32);
        uint32_t b = (uint32_t)__shfl_xor((int)kmax, off, 32);
        kmin = kmin < a ? kmin : a;
        kmax = kmax > b ? kmax : b;
    }
    if ((tid & 31u) == 0u) {
        atomicMin(&minmax[0], kmin);
        atomicMax(&minmax[1], kmax);
    }
}

// ------------------- finalize: affine z-normalization + avg + row flip ------
// mean of normalized z == 255*(mean_raw - zmin)/range; apply per pixel while
// swapping row y with row H-1-y in place (sums live in d_out).
__global__ void finalize(float* __restrict__ out, const float* __restrict__ counts,
                         const uint32_t* __restrict__ minmax) {
    int i = blockIdx.x * blockDim.x + threadIdx.x;   // i over top-half pixels
    if (i >= (H / 2) * W) return;
    float zmin = key2f(minmax[0]);
    float zmax = key2f(minmax[1]);
    float zscale = 255.0f / (zmax - zmin);

    int y = i / W, x = i % W;
    int top = y * W + x;
    int bot = (H - 1 - y) * W + x;
    float sT = out[top], cT = counts[top];
    float sB = out[bot], cB = counts[bot];
    out[top] = (cB > 0.0f) ? zscale * (sB / fmaxf(cB, 1.0f) - zmin) : 0.0f;
    out[bot] = (cT > 0.0f) ? zscale * (sT / fmaxf(cT, 1.0f) - zmin) : 0.0f;
}

// ------------------------------------------------------------------------
extern "C" void kernel_launch(void* const* d_in, const int* in_sizes, int n_in,
                              void* d_out, int out_size, void* d_ws, size_t ws_size,
                              hipStream_t stream) {
    const float* pc = (const float*)d_in[0];
    const float* C7 = (const float*)d_in[1];
    const float* DR = (const float*)d_in[2];
    const float* DL = (const float*)d_in[3];
    long long n = (long long)in_sizes[0] / 3;

    uint8_t*  ws     = (uint8_t*)d_ws;
    uint32_t* minmax = (uint32_t*)ws;            // 8 bytes
    float*    counts = (float*)(ws + 256);       // NPIX floats (1 MB)
    float*    sums   = (float*)d_out;            // accumulate raw-z sums in d_out

    init_ws<<<512, 256, 0, stream>>>(minmax, sums, counts);
    bin_points<<<2048, BLOCK, 0, stream>>>(pc, n, C7, DR, DL, minmax, sums, counts);
    finalize<<<((H / 2) * W + 255) / 256, 256, 0, stream>>>(sums, counts, minmax);
}